// TemporalGraphModule_66245575573973
// MI455X (gfx1250) — compile-verified
//
#include <hip/hip_runtime.h>
#include <math.h>

// ---------------------------------------------------------------------------
// Problem constants: B=8, N=256 (seq), F=256, K=2, H=128, 4H=512, 2H=256
// ---------------------------------------------------------------------------
typedef __attribute__((ext_vector_type(2))) float v2f;
typedef __attribute__((ext_vector_type(8))) float v8f;

#define TG_CONST 0.00390625f   // exact fp32 value of softmax(const row) = 1/256

// ---------------------------------------------------------------------------
// TG is exactly constant (LayerNorm over a size-1 dim outputs its bias; the
// row softmax of a constant row is 1/N; 256 + 1e-8 rounds to 256.0 in fp32).
// ---------------------------------------------------------------------------
__global__ __launch_bounds__(256) void fill_tg_kernel(float* __restrict__ tg) {
  size_t i = (size_t)blockIdx.x * 256u + threadIdx.x;
  tg[i] = TG_CONST;
}

// xsum[b, f] = sum_n x[b, n, f]
__global__ __launch_bounds__(256) void colsum_kernel(const float* __restrict__ x,
                                                     float* __restrict__ xsum) {
  int b = blockIdx.x, o = threadIdx.x;
  const float* xb = x + (size_t)b * 256 * 256;
  float s = 0.f;
  for (int n = 0; n < 256; ++n) s += xb[(size_t)n * 256 + o];
  xsum[b * 256 + o] = s;
}

// Wc = W0 + 0.5*W1   (folds P's identity part: out = x@Wc + (1/512)*xsum@W1)
__global__ __launch_bounds__(256) void combine_w_kernel(const float* __restrict__ w0,
                                                        const float* __restrict__ w1,
                                                        float* __restrict__ wc) {
  int i = blockIdx.x * 256 + threadIdx.x;
  wc[i] = w0[i] + 0.5f * w1[i];
}

// bias2[b, o] = cheb_b[o] + (1/512) * sum_f xsum[b,f] * W1[f,o]
__global__ __launch_bounds__(256) void bias2_kernel(const float* __restrict__ xsum,
                                                    const float* __restrict__ w1,
                                                    const float* __restrict__ cheb_b,
                                                    float* __restrict__ bias2) {
  int b = blockIdx.x, o = threadIdx.x;
  float acc = 0.f;
  for (int f = 0; f < 256; ++f) acc += xsum[b * 256 + f] * w1[(size_t)f * 256 + o];
  bias2[b * 256 + o] = cheb_b[o] + acc * (1.f / 512.f);
}

// ---------------------------------------------------------------------------
// WMMA GEMM (fp32, 16x16x4): C[2048,256] = A[2048,256] @ Bm[256,256] + bias2[row>>8][n]
// One 16x16 output tile per wave; 8 waves per block.
// ---------------------------------------------------------------------------
__global__ __launch_bounds__(256) void cheb_gemm_wmma(const float* __restrict__ A,
                                                      const float* __restrict__ Bm,
                                                      const float* __restrict__ bias2,
                                                      float* __restrict__ C) {
  const int N = 256, K = 256;
  int wave = threadIdx.x >> 5;
  int lane = threadIdx.x & 31;
  int tile = blockIdx.x * 8 + wave;
  int tm = tile >> 4;            // 16 N-tiles
  int tn = tile & 15;
  int m0 = tm * 16, n0 = tn * 16;
  int ml = lane & 15;
  int koff = (lane < 16) ? 0 : 2;
  const float* arow = A + (size_t)(m0 + ml) * K + koff;
  int ncol = n0 + ml;
  v8f acc = {};
  for (int kg = 0; kg < K / 4; ++kg) {
    v2f a = *(const v2f*)(arow + kg * 4);
    int kb = kg * 4 + koff;
    v2f b;
    b.x = Bm[(size_t)kb * N + ncol];
    b.y = Bm[(size_t)(kb + 1) * N + ncol];
    acc = __builtin_amdgcn_wmma_f32_16x16x4_f32(false, a, false, b, (short)0, acc,
                                                false, false);
  }
  int mbase = m0 + ((lane < 16) ? 0 : 8);
#pragma unroll
  for (int r = 0; r < 8; ++r) {
    int m = mbase + r;
    C[(size_t)m * N + ncol] = acc[r] + bias2[(m >> 8) * 256 + ncol];
  }
}

// ---------------------------------------------------------------------------
// WMMA GEMM (fp32): Gin[2048,512] = A[2048,K=256] @ Wih[512,256]^T + (bih+bhh)[n]
// B is row-major [N,K] -> contiguous float2 loads per lane.
// ---------------------------------------------------------------------------
__global__ __launch_bounds__(256) void gin_gemm_wmma(const float* __restrict__ A,
                                                     const float* __restrict__ W,
                                                     const float* __restrict__ bih,
                                                     const float* __restrict__ bhh,
                                                     float* __restrict__ C) {
  const int N = 512, K = 256;
  int wave = threadIdx.x >> 5;
  int lane = threadIdx.x & 31;
  int tile = blockIdx.x * 8 + wave;
  int tm = tile >> 5;            // 32 N-tiles
  int tn = tile & 31;
  int m0 = tm * 16, n0 = tn * 16;
  int ml = lane & 15;
  int koff = (lane < 16) ? 0 : 2;
  const float* arow = A + (size_t)(m0 + ml) * K + koff;
  int ncol = n0 + ml;
  const float* brow = W + (size_t)ncol * K + koff;
  v8f acc = {};
  for (int kg = 0; kg < K / 4; ++kg) {
    v2f a = *(const v2f*)(arow + kg * 4);
    v2f b = *(const v2f*)(brow + kg * 4);
    acc = __builtin_amdgcn_wmma_f32_16x16x4_f32(false, a, false, b, (short)0, acc,
                                                false, false);
  }
  float bias = bih[ncol] + bhh[ncol];
  int mbase = m0 + ((lane < 16) ? 0 : 8);
#pragma unroll
  for (int r = 0; r < 8; ++r) {
    C[(size_t)(mbase + r) * N + ncol] = acc[r] + bias;
  }
}

// ---------------------------------------------------------------------------
// Persistent LSTM recurrence (one workgroup per direction, 32 waves).
// Per step: R[8,512] = h[8,128] @ Whh^T via WMMA (M padded to 16); each wave
// owns one 16-wide gate column tile and caches its Whh^T fragments in VGPRs
// for all 256 timesteps. Then 1024 threads do the i,f,g,o pointwise update.
//   Gin: [dir][b, t, 512] (bias pre-added)   Whh: [dir][512, 128]
//   outbuf: [b, t, 256], dir d writes columns [128*d, 128*d+128)
// ---------------------------------------------------------------------------
__global__ __launch_bounds__(1024) void lstm_rec_kernel(const float* __restrict__ Gin,
                                                        const float* __restrict__ Whh,
                                                        float* __restrict__ outbuf) {
  const int dir = blockIdx.x;
  const float* gin = Gin + (size_t)dir * (8 * 256 * 512);
  const float* whh = Whh + (size_t)dir * (512 * 128);

  __shared__ float hsh[8][132];
  __shared__ float csh[8][132];
  __shared__ float gsh[8][520];

  const int tid = threadIdx.x;
  const int wave = tid >> 5;
  const int lane = tid & 31;
  const int eb = tid >> 7;        // elementwise batch  0..7
  const int eu = tid & 127;       // elementwise hidden 0..127

  hsh[eb][eu] = 0.f;
  csh[eb][eu] = 0.f;

  // Cache this wave's B fragments (Whh^T tile, N columns [16*wave, 16*wave+16))
  const int n0 = wave * 16;
  const int ml = lane & 15;
  const int koff = (lane < 16) ? 0 : 2;
  const bool mvalid = ml < 8;     // rows 8..15 of the 16-row A tile are padding
  const float* brow = whh + (size_t)(n0 + ml) * 128 + koff;
  v2f bfrag[32];
#pragma unroll
  for (int kg = 0; kg < 32; ++kg) bfrag[kg] = *(const v2f*)(brow + kg * 4);

  __syncthreads();

  for (int step = 0; step < 256; ++step) {
    const int t = dir ? (255 - step) : step;

    // ---- R = h @ Whh^T for this wave's 16 gate columns ----
    v8f acc = {};
#pragma unroll
    for (int kg = 0; kg < 32; ++kg) {
      v2f a = {0.f, 0.f};
      if (mvalid) a = *(const v2f*)(&hsh[ml][kg * 4 + koff]);
      acc = __builtin_amdgcn_wmma_f32_16x16x4_f32(false, a, false, bfrag[kg],
                                                  (short)0, acc, false, false);
    }
    // lanes 0..15 hold rows m=0..7 (real batches); lanes 16..31 hold padding
    if (lane < 16) {
#pragma unroll
      for (int r = 0; r < 8; ++r) gsh[r][n0 + lane] = acc[r];
    }
    __syncthreads();

    // ---- pointwise gate update: one thread per (b, u) ----
    {
      const float* gi = gin + ((size_t)(eb * 256 + t)) * 512;
      float xi = gsh[eb][eu]       + gi[eu];
      float xf = gsh[eb][128 + eu] + gi[128 + eu];
      float xg = gsh[eb][256 + eu] + gi[256 + eu];
      float xo = gsh[eb][384 + eu] + gi[384 + eu];
      float si = 1.f / (1.f + expf(-xi));
      float sf = 1.f / (1.f + expf(-xf));
      float so = 1.f / (1.f + expf(-xo));
      float c  = sf * csh[eb][eu] + si * tanhf(xg);
      float h  = so * tanhf(c);
      csh[eb][eu] = c;
      hsh[eb][eu] = h;
      outbuf[((size_t)(eb * 256 + t)) * 256 + dir * 128 + eu] = h;
    }
    __syncthreads();
  }
}

// ---------------------------------------------------------------------------
// Row LayerNorm over 256 features: out = (x - m) * rsqrt(v + 1e-5) * g + b
// ---------------------------------------------------------------------------
__global__ __launch_bounds__(256) void ln256_kernel(const float* __restrict__ in,
                                                    const float* __restrict__ g,
                                                    const float* __restrict__ b,
                                                    float* __restrict__ out) {
  __shared__ float red[256];
  const int row = blockIdx.x, tid = threadIdx.x;
  float v = in[(size_t)row * 256 + tid];
  red[tid] = v;
  __syncthreads();
  for (int s = 128; s > 0; s >>= 1) {
    if (tid < s) red[tid] += red[tid + s];
    __syncthreads();
  }
  float mean = red[0] * (1.f / 256.f);
  __syncthreads();
  float d = v - mean;
  red[tid] = d * d;
  __syncthreads();
  for (int s = 128; s > 0; s >>= 1) {
    if (tid < s) red[tid] += red[tid + s];
    __syncthreads();
  }
  float var = red[0] * (1.f / 256.f);
  out[(size_t)row * 256 + tid] = d * rsqrtf(var + 1e-5f) * g[tid] + b[tid];
}

// agg[b, o] = mean_t lstm_out[b, t, o]
__global__ __launch_bounds__(256) void agg_kernel(const float* __restrict__ lstm_out,
                                                  float* __restrict__ agg) {
  int b = blockIdx.x, o = threadIdx.x;
  float s = 0.f;
  for (int t = 0; t < 256; ++t) s += lstm_out[((size_t)(b * 256 + t)) * 256 + o];
  agg[b * 256 + o] = s * (1.f / 256.f);
}

// ---------------------------------------------------------------------------
extern "C" void kernel_launch(void* const* d_in, const int* in_sizes, int n_in,
                              void* d_out, int out_size, void* d_ws, size_t ws_size,
                              hipStream_t stream) {
  (void)in_sizes; (void)n_in; (void)out_size; (void)ws_size;

  const float* x         = (const float*)d_in[0];   // [8,256,256]
  // d_in[1..4] (w_adj, b_adj, ln_adj_g, ln_adj_b) are provably unused:
  // LayerNorm(1) forces the adjacency logits to a constant.
  const float* cheb_w    = (const float*)d_in[5];   // [2,256,256]
  const float* cheb_b    = (const float*)d_in[6];   // [256]
  const float* ln_cheb_g = (const float*)d_in[7];
  const float* ln_cheb_b = (const float*)d_in[8];
  const float* w_ih      = (const float*)d_in[9];   // [2,2,512,256]
  const float* w_hh      = (const float*)d_in[10];  // [2,2,512,128]
  const float* b_ih      = (const float*)d_in[11];  // [2,2,512]
  const float* b_hh      = (const float*)d_in[12];  // [2,2,512]
  const float* ln_out_g  = (const float*)d_in[13];  // [256]
  const float* ln_out_b  = (const float*)d_in[14];  // [256]

  float* out      = (float*)d_out;
  float* agg      = out;                  // [8,256]        = 2048
  float* tg       = out + 2048;           // [8,256,256]    = 524288
  float* lstm_out = out + 2048 + 524288;  // [8,256,256]    = 524288

  float* ws       = (float*)d_ws;
  float* xsum     = ws;                   // 2048
  float* bias2    = ws + 2048;            // 2048
  float* Wc       = ws + 4096;            // 65536
  float* cheb_tmp = ws + 69632;           // 524288  (reused as layer-0 hidden seq)
  float* cheb_out = ws + 593920;          // 524288  (reused as layer-1 hidden seq)
  float* Gin      = ws + 1118208;         // 2 x 1048576 = [dir][2048,512]

  // TG is an exact constant
  fill_tg_kernel<<<2048, 256, 0, stream>>>(tg);

  // ChebConv collapsed: out = x @ (W0 + 0.5 W1) + (1/512) xsum @ W1 + cheb_b
  colsum_kernel<<<8, 256, 0, stream>>>(x, xsum);
  combine_w_kernel<<<256, 256, 0, stream>>>(cheb_w, cheb_w + 65536, Wc);
  bias2_kernel<<<8, 256, 0, stream>>>(xsum, cheb_w + 65536, cheb_b, bias2);
  cheb_gemm_wmma<<<256, 256, 0, stream>>>(x, Wc, bias2, cheb_tmp);
  ln256_kernel<<<2048, 256, 0, stream>>>(cheb_tmp, ln_cheb_g, ln_cheb_b, cheb_out);

  // ---- BiLSTM layer 0: precompute input projections, then recurrence ----
  gin_gemm_wmma<<<512, 256, 0, stream>>>(cheb_out, w_ih + 0,      b_ih + 0,
                                         b_hh + 0,   Gin);
  gin_gemm_wmma<<<512, 256, 0, stream>>>(cheb_out, w_ih + 131072, b_ih + 512,
                                         b_hh + 512, Gin + 1048576);
  lstm_rec_kernel<<<2, 1024, 0, stream>>>(Gin, w_hh + 0, cheb_tmp);  // -> hseq0

  // ---- BiLSTM layer 1 ----
  gin_gemm_wmma<<<512, 256, 0, stream>>>(cheb_tmp, w_ih + 262144, b_ih + 1024,
                                         b_hh + 1024, Gin);
  gin_gemm_wmma<<<512, 256, 0, stream>>>(cheb_tmp, w_ih + 393216, b_ih + 1536,
                                         b_hh + 1536, Gin + 1048576);
  lstm_rec_kernel<<<2, 1024, 0, stream>>>(Gin, w_hh + 131072, cheb_out);  // -> hseq1

  // ---- Output LayerNorm + mean pooling ----
  ln256_kernel<<<2048, 256, 0, stream>>>(cheb_out, ln_out_g, ln_out_b, lstm_out);
  agg_kernel<<<8, 256, 0, stream>>>(lstm_out, agg);
}